// quan_Linear_fi_30889404793388
// MI455X (gfx1250) — compile-verified
//
#include <hip/hip_runtime.h>
#include <stdint.h>

#define B_    2
#define S_    128
#define INF_  512
#define OUTF_ 512
#define QMAXW 127.0f     // 2^(8-1)-1
#define QMAXP 32767.0f   // 2^(16-1)-1
#define HALF_RANGE 32768

typedef float v2f __attribute__((ext_vector_type(2)));
typedef float v8f __attribute__((ext_vector_type(8)));

// workspace float layout
#define WS_MAXW 0
#define WS_MAXP 1
#define WS_MAXC 2
#define WS_MAXY 3
#define WS_WQ   8
#define WS_WQT  (WS_WQ  + OUTF_*INF_)
#define WS_Y0   (WS_WQT + OUTF_*INF_)

// stream salts (element indices fit in 26 bits -> 32-bit hashing is enough)
#define SALT_P  0x00000000u
#define SALT_C  0x9E3779B9u
#define SALT_Y  0x85EBCA6Bu

// lowbias32 hash: ~6 VALU ops, full 32-bit avalanche
__device__ __forceinline__ uint32_t hash32(uint32_t x) {
  x ^= x >> 16; x *= 0x7feb352dU;
  x ^= x >> 15; x *= 0x846ca68bU;
  x ^= x >> 16;
  return x;
}

// Single-draw approximation of 16 independent Bernoulli(1e-5) bit flips:
// P(any flip) = 1-(1-1e-5)^16 ~ 1.59988e-4 -> threshold 2684 on 24 bits.
// (P(two flips) ~ 1e-8: negligible.) Returns dequantized flip error for the
// offset-binary code of qi. Caller keeps this on the rare path.
__device__ __forceinline__ float flip_err(int qi, uint32_t h, float scale) {
  int k = (h >> 24) & 15;
  uint32_t u = (uint32_t)(qi + HALF_RANGE);
  float mag = (float)(1 << k) * scale;
  return ((u >> k) & 1u) ? -mag : mag;
}

__device__ __forceinline__ bool flips(uint32_t h) {
  return (h & 0xFFFFFFu) < 2684u;
}

__device__ __forceinline__ void blockMaxToGlobal(float v, float* slot) {
  __shared__ float red[256];
  red[threadIdx.x] = v;
  __syncthreads();
  for (int s = blockDim.x >> 1; s > 0; s >>= 1) {
    if ((int)threadIdx.x < s) red[threadIdx.x] = fmaxf(red[threadIdx.x], red[threadIdx.x + s]);
    __syncthreads();
  }
  if (threadIdx.x == 0)
    atomicMax((unsigned int*)slot, __float_as_uint(red[0]));  // values are >= 0
}

__global__ void k_init(float* ws) {
  if (threadIdx.x < 8) ws[threadIdx.x] = 0.0f;
}

__global__ void k_maxabs_w(const float* __restrict__ w, float* ws) {
  float m = 0.0f;
  for (int i = blockIdx.x * blockDim.x + threadIdx.x; i < OUTF_ * INF_;
       i += gridDim.x * blockDim.x)
    m = fmaxf(m, fabsf(w[i]));
  blockMaxToGlobal(m, &ws[WS_MAXW]);
}

__global__ void k_quant_w(const float* __restrict__ w, float* ws) {
  int idx = blockIdx.x * blockDim.x + threadIdx.x;
  if (idx >= OUTF_ * INF_) return;
  float step = ws[WS_MAXW] / QMAXW;
  float inv  = step > 0.0f ? 1.0f / step : 0.0f;
  float q = fminf(fmaxf(rintf(w[idx] * inv), -128.0f), 127.0f);
  float wq = q * step;
  int o = idx / INF_, i = idx - o * INF_;
  ws[WS_WQ  + idx]            = wq;   // [OUT, IN] for WMMA B-fragments
  ws[WS_WQT + i * OUTF_ + o]  = wq;   // [IN, OUT] for coalesced scan reads
}

// max|p| = max_i ( max_{b,s}|x[:,:,i]| * max_o|wq[o,i]| )  -- exactly separable
__global__ void k_maxp(const float* __restrict__ x, float* ws) {
  int i = blockIdx.x * blockDim.x + threadIdx.x;
  if (i >= INF_) return;
  float mx = 0.0f;
  for (int r = 0; r < B_ * S_; ++r) mx = fmaxf(mx, fabsf(x[r * INF_ + i]));
  float mw = 0.0f;
  const float* wqt = ws + WS_WQT + (size_t)i * OUTF_;
  for (int o = 0; o < OUTF_; ++o) mw = fmaxf(mw, fabsf(wqt[o]));
  atomicMax((unsigned int*)&ws[WS_MAXP], __float_as_uint(mx * mw));
}

// y0[b,s,o] = sum_i x[b,s,i]*wq[o,i] via V_WMMA_F32_16X16X4_F32 (exact f32).
// One 16x16 output tile per wave32. A = x tile (M=s,K=i), B = wq^T tile (K=i,N=o).
// f32 16x4 A layout: lanes 0-15 hold {K=0,K=1}, lanes 16-31 hold {K=2,K=3} -> float2 loads.
__global__ void k_gemm_wmma(const float* __restrict__ x, float* ws) {
  const float* wq = ws + WS_WQ;
  float*       y0 = ws + WS_Y0;
  int wave = blockIdx.x * (blockDim.x >> 5) + (threadIdx.x >> 5);
  int lane = threadIdx.x & 31;
  int ot = wave & 31;          // OUT/16 = 32 tiles
  int st = (wave >> 5) & 7;    // S/16   = 8 tiles
  int b  = wave >> 8;          // B      = 2
  int o0 = ot * 16, s0 = st * 16;
  int half = lane >> 4, r = lane & 15;

  const float* xrow = x  + (size_t)(b * S_ + s0 + r) * INF_ + 2 * half;
  const float* wrow = wq + (size_t)(o0 + r)          * INF_ + 2 * half;

  v8f acc0 = {0.f, 0.f, 0.f, 0.f, 0.f, 0.f, 0.f, 0.f};
  v8f acc1 = {0.f, 0.f, 0.f, 0.f, 0.f, 0.f, 0.f, 0.f};

#pragma unroll 4
  for (int k = 0; k < INF_; k += 8) {
    __builtin_prefetch(xrow + k + 64, 0, 0);
    __builtin_prefetch(wrow + k + 64, 0, 0);
    v2f a0 = *(const v2f*)(xrow + k);
    v2f b0 = *(const v2f*)(wrow + k);
    acc0 = __builtin_amdgcn_wmma_f32_16x16x4_f32(
        false, a0, false, b0, (short)0, acc0, false, false);
    v2f a1 = *(const v2f*)(xrow + k + 4);
    v2f b1 = *(const v2f*)(wrow + k + 4);
    acc1 = __builtin_amdgcn_wmma_f32_16x16x4_f32(
        false, a1, false, b1, (short)0, acc1, false, false);
  }
  v8f acc = acc0 + acc1;

  // D layout: VGPR j -> M = j + 8*half, N = lane&15
#pragma unroll
  for (int j = 0; j < 8; ++j) {
    int srow = s0 + j + 8 * half;
    y0[(size_t)(b * S_ + srow) * OUTF_ + (o0 + r)] = acc[j];
  }
}

// Scan kernels: one thread per (b,s,o). (b,s) is made explicitly block-uniform
// (bs = blockIdx>>1) so x-row reads become scalar (constant-cache) loads;
// wqT reads are lane-coalesced and L2-resident.

// Pass 1: running c = cumsum(quant16(p)+flip); record max|c|.
__global__ void k_scan_maxc(const float* __restrict__ x, float* ws) {
  int bs = blockIdx.x >> 1;                              // uniform per block
  int o  = ((blockIdx.x & 1) << 8) | threadIdx.x;        // 0..511
  int s = bs & (S_ - 1), b = bs >> 7;
  const float* xrow = x + (size_t)bs * INF_;
  const float* wqt  = ws + WS_WQT + o;
  float sp = ws[WS_MAXP] / QMAXP; sp = sp > 0.0f ? sp : 1.0f;
  float isp = 1.0f / sp;
  uint32_t ebase = (uint32_t)((b * OUTF_ + o) * S_ + s) * (uint32_t)INF_;

  float c = 0.0f, mc = 0.0f;
#pragma unroll 4
  for (int i = 0; i < INF_; ++i) {
    float p  = xrow[i] * wqt[(size_t)i * OUTF_];
    float qf = fminf(fmaxf(rintf(p * isp), -32768.0f), 32767.0f);
    float pq = qf * sp;
    uint32_t hp = hash32((ebase + i) + SALT_P);
    if (__builtin_expect(flips(hp), 0)) pq += flip_err((int)qf, hp, sp);
    c += pq;
    mc = fmaxf(mc, fabsf(c));
  }
  blockMaxToGlobal(mc, &ws[WS_MAXC]);
}

// Pass 2: same scan (same hashes -> same flips), accumulate quant+flip
// corrections; y = y0 + corrections; record max|y|.
__global__ void k_scan_corr(const float* __restrict__ x, float* ws) {
  int bs = blockIdx.x >> 1;
  int o  = ((blockIdx.x & 1) << 8) | threadIdx.x;
  int s = bs & (S_ - 1), b = bs >> 7;
  int gid = (bs << 9) | o;
  const float* xrow = x + (size_t)bs * INF_;
  const float* wqt  = ws + WS_WQT + o;
  float sp = ws[WS_MAXP] / QMAXP; sp = sp > 0.0f ? sp : 1.0f;
  float sc = ws[WS_MAXC] / QMAXP; sc = sc > 0.0f ? sc : 1.0f;
  float isp = 1.0f / sp, isc = 1.0f / sc;
  uint32_t ebase = (uint32_t)((b * OUTF_ + o) * S_ + s) * (uint32_t)INF_;

  float c = 0.0f, perr = 0.0f, cerr = 0.0f;
#pragma unroll 4
  for (int i = 0; i < INF_; ++i) {
    float p  = xrow[i] * wqt[(size_t)i * OUTF_];
    float qf = fminf(fmaxf(rintf(p * isp), -32768.0f), 32767.0f);
    float pq = qf * sp;
    uint32_t hp = hash32((ebase + i) + SALT_P);
    if (__builtin_expect(flips(hp), 0)) pq += flip_err((int)qf, hp, sp);
    perr += (pq - p);                 // p-stage: quantization + flip error
    c += pq;
    float qc = fminf(fmaxf(rintf(c * isc), -32768.0f), 32767.0f);
    float e  = qc * sc - c;
    uint32_t hc = hash32((ebase + i) + SALT_C);
    if (__builtin_expect(flips(hc), 0)) e += flip_err((int)qc, hc, sc);
    if (i >= 1) cerr += e;            // j=0 excluded; j=IN-1 enters via y_sum
  }
  float y = ws[WS_Y0 + gid] + perr + cerr;
  ws[WS_Y0 + gid] = y;
  blockMaxToGlobal(fabsf(y), &ws[WS_MAXY]);
}

// Final 16-bit quant+flip of y, transpose is implicit (y0 stored [B,S,OUT]), add bias.
__global__ void k_final(const float* __restrict__ bias, const float* __restrict__ ws,
                        float* __restrict__ out) {
  int gid = blockIdx.x * blockDim.x + threadIdx.x;
  int o  = gid & (OUTF_ - 1);
  int bs = gid >> 9;
  int s = bs & (S_ - 1), b = bs >> 7;
  float sy = ws[WS_MAXY] / QMAXP; sy = sy > 0.0f ? sy : 1.0f;
  float y  = ws[WS_Y0 + gid];
  float qf = fminf(fmaxf(rintf(y / sy), -32768.0f), 32767.0f);
  float yq = qf * sy;
  uint32_t hy = hash32((uint32_t)((b * OUTF_ + o) * S_ + s) + SALT_Y);
  if (__builtin_expect(flips(hy), 0)) yq += flip_err((int)qf, hy, sy);
  out[gid] = yq + bias[o];
}

extern "C" void kernel_launch(void* const* d_in, const int* in_sizes, int n_in,
                              void* d_out, int out_size, void* d_ws, size_t ws_size,
                              hipStream_t stream) {
  const float* x    = (const float*)d_in[0];
  const float* w    = (const float*)d_in[1];
  const float* bias = (const float*)d_in[2];
  float* out = (float*)d_out;
  float* ws  = (float*)d_ws;
  (void)in_sizes; (void)n_in; (void)out_size; (void)ws_size;

  k_init<<<1, 32, 0, stream>>>(ws);
  k_maxabs_w<<<256, 256, 0, stream>>>(w, ws);
  k_quant_w<<<(OUTF_ * INF_ + 255) / 256, 256, 0, stream>>>(w, ws);
  k_maxp<<<(INF_ + 255) / 256, 256, 0, stream>>>(x, ws);
  k_gemm_wmma<<<128, 128, 0, stream>>>(x, ws);          // 512 waves = 512 tiles
  k_scan_maxc<<<512, 256, 0, stream>>>(x, ws);          // block-uniform (b,s)
  k_scan_corr<<<512, 256, 0, stream>>>(x, ws);
  k_final<<<512, 256, 0, stream>>>(bias, ws, out);
}